// WarpRenderer_62861141344857
// MI455X (gfx1250) — compile-verified
//
#include <hip/hip_runtime.h>
#include <math.h>

#ifndef __has_builtin
#define __has_builtin(x) 0
#endif

#if __has_builtin(__builtin_amdgcn_global_load_async_to_lds_b128)
#define HAVE_ASYNC_LDS 1
#endif

static constexpr int B  = 4;
static constexpr int NI = 16;
static constexpr int C  = 3;
static constexpr int H  = 256;
static constexpr int W  = 256;
static constexpr int HW = H * W;

typedef int v4i __attribute__((vector_size(16)));
typedef __attribute__((address_space(1))) v4i* gv4i_ptr;
typedef __attribute__((address_space(3))) v4i* lv4i_ptr;

// ---------------------------------------------------------------------------
// Prep: per-(b,ni) argmax over 5 mode logits -> scalar coefficient, plus theta.
// table layout: [B*NI][8] = {t00,t01,t02,t10,t11,t12,coeff,pad}
// ---------------------------------------------------------------------------
__global__ __launch_bounds__(64) void warp_prep_kernel(
    const float* __restrict__ mode_logits,
    const float* __restrict__ transform_params,
    const float* __restrict__ alpha,
    const float* __restrict__ instance_valid,
    float* __restrict__ table)
{
    int t = threadIdx.x;
    if (t >= B * NI) return;
    const float* lg = mode_logits + t * 5;
    int best = 0;
    float bv = lg[0];
#pragma unroll
    for (int j = 1; j < 5; ++j) {
        float v = lg[j];
        if (v > bv) { bv = v; best = j; }   // first-max wins, matches jnp.argmax
    }
    float stat  = (best == 0 || best == 2 || best == 4) ? 1.f : 0.f;
    float blink = (best == 3) ? 1.f : 0.f;
    float coeff = (stat + blink * alpha[t]) * instance_valid[t];
    const float* p = transform_params + t * 6;
    float* o = table + t * 8;
    o[0] = p[0]; o[1] = p[1]; o[2] = p[2];
    o[3] = p[3]; o[4] = p[4]; o[5] = p[5];
    o[6] = coeff; o[7] = 0.f;
}

// ---------------------------------------------------------------------------
// Main fused kernel: one thread = one (b, y, x); loops over NI instances.
// Writes g_mid_per_instance, m_mid_per_instance (nontemporal, streaming),
// accumulates g_mid / m_mid in registers, then writes g_mid, m_mid, i_base.
// ---------------------------------------------------------------------------
__global__ __launch_bounds__(256) void warp_render_kernel(
    const float* __restrict__ g0,
    const float* __restrict__ m0,
    const float* __restrict__ i_bg,
    const float* __restrict__ table,
    float* __restrict__ out_gpi,
    float* __restrict__ out_mpi,
    float* __restrict__ out_gmid,
    float* __restrict__ out_mmid,
    float* __restrict__ out_ibase)
{
    __shared__ float tab[NI * 8];
    const int tid = threadIdx.x;
    const int p   = blockIdx.x * 256 + tid;
    const int b   = p >> 16;            // HW == 65536
    const int rem = p & (HW - 1);
    const int y   = rem >> 8;
    const int x   = rem & (W - 1);

    // ---- stage this batch's 16x8 coefficient/theta table into LDS ----------
    const float* gt = table + b * NI * 8;
#if defined(HAVE_ASYNC_LDS)
    if (tid < (NI * 8 * 4) / 16) {      // 32 lanes x 16B = 512B, one b128 async op
        const float* gsrc = gt + tid * 4;
        float* ldst = &tab[tid * 4];
        __builtin_amdgcn_global_load_async_to_lds_b128(
            (gv4i_ptr)gsrc, (lv4i_ptr)ldst, 0, 0);
#if __has_builtin(__builtin_amdgcn_s_wait_asynccnt)
        __builtin_amdgcn_s_wait_asynccnt(0);
#else
        asm volatile("s_wait_asynccnt 0" ::: "memory");
#endif
    }
#else
    if (tid < NI * 8) tab[tid] = gt[tid];
#endif
    __syncthreads();

    // ---- prefetch the tail i_bg reads (used only after the NI loop) --------
    const int bo = b * C * HW + rem;
    __builtin_prefetch(&i_bg[bo], 0, 1);
    __builtin_prefetch(&i_bg[bo + HW], 0, 1);
    __builtin_prefetch(&i_bg[bo + 2 * HW], 0, 1);

    // normalized output-grid coords (align_corners=False convention)
    const float xs = (2.f * (float)x + 1.f) * (1.f / (float)W) - 1.f;
    const float ys = (2.f * (float)y + 1.f) * (1.f / (float)H) - 1.f;

    float accr = 0.f, accg = 0.f, accb = 0.f, accm = 0.f;

    for (int ni = 0; ni < NI; ++ni) {
        const float* tt = &tab[ni * 8];
        // block-uniform coefficient -> scalar branch for dead instances
        float coeff = __int_as_float(
            __builtin_amdgcn_readfirstlane(__float_as_int(tt[6])));

        float* gp = out_gpi + (size_t)(b * NI + ni) * C * HW + rem;
        float* mp = out_mpi + (size_t)(b * NI + ni) * HW + rem;

        if (coeff == 0.f) {             // invalid instance or mode==1
            __builtin_nontemporal_store(0.f, gp);
            __builtin_nontemporal_store(0.f, gp + HW);
            __builtin_nontemporal_store(0.f, gp + 2 * HW);
            __builtin_nontemporal_store(0.f, mp);
            continue;
        }

        float gx = tt[0] * xs + tt[1] * ys + tt[2];
        float gy = tt[3] * xs + tt[4] * ys + tt[5];
        float ix = ((gx + 1.f) * (float)W - 1.f) * 0.5f;
        float iy = ((gy + 1.f) * (float)H - 1.f) * 0.5f;
        float x0f = floorf(ix), y0f = floorf(iy);
        float wx1 = ix - x0f, wy1 = iy - y0f;
        float wx0 = 1.f - wx1, wy0 = 1.f - wy1;
        int x0 = (int)x0f, y0 = (int)y0f;
        int x1 = x0 + 1,   y1 = y0 + 1;

        float mx0 = (x0 >= 0 && x0 < W) ? 1.f : 0.f;
        float mx1 = (x1 >= 0 && x1 < W) ? 1.f : 0.f;
        float my0 = (y0 >= 0 && y0 < H) ? 1.f : 0.f;
        float my1 = (y1 >= 0 && y1 < H) ? 1.f : 0.f;

        int x0c = min(max(x0, 0), W - 1);
        int x1c = min(max(x1, 0), W - 1);
        int y0c = min(max(y0, 0), H - 1);
        int y1c = min(max(y1, 0), H - 1);
        int i00 = y0c * W + x0c;
        int i10 = y0c * W + x1c;
        int i01 = y1c * W + x0c;
        int i11 = y1c * W + x1c;

        float w00 = wx0 * wy0 * (mx0 * my0);
        float w10 = wx1 * wy0 * (mx1 * my0);
        float w01 = wx0 * wy1 * (mx0 * my1);
        float w11 = wx1 * wy1 * (mx1 * my1);

        const float* src = g0 + (size_t)(b * NI + ni) * C * HW;
        const float* srm = m0 + (size_t)(b * NI + ni) * HW;

        // mask channel
        float vm = w00 * srm[i00] + w10 * srm[i10]
                 + w01 * srm[i01] + w11 * srm[i11];
        vm *= coeff;
        __builtin_nontemporal_store(vm, mp);
        accm += vm;

        // 3 color channels
#pragma unroll
        for (int c = 0; c < C; ++c) {
            const float* sc = src + c * HW;
            float v = w00 * sc[i00] + w10 * sc[i10]
                    + w01 * sc[i01] + w11 * sc[i11];
            v *= coeff;
            __builtin_nontemporal_store(v, gp + c * HW);
            if (c == 0) accr += v; else if (c == 1) accg += v; else accb += v;
        }
    }

    // ---- reductions + composite -------------------------------------------
    out_gmid[bo]           = accr;
    out_gmid[bo + HW]      = accg;
    out_gmid[bo + 2 * HW]  = accb;
    out_mmid[b * HW + rem] = fminf(fmaxf(accm, 0.f), 1.f);
    out_ibase[bo]          = i_bg[bo]          + accr;
    out_ibase[bo + HW]     = i_bg[bo + HW]     + accg;
    out_ibase[bo + 2 * HW] = i_bg[bo + 2 * HW] + accb;
}

extern "C" void kernel_launch(void* const* d_in, const int* in_sizes, int n_in,
                              void* d_out, int out_size, void* d_ws, size_t ws_size,
                              hipStream_t stream) {
    (void)in_sizes; (void)n_in; (void)out_size; (void)ws_size;
    const float* g0  = (const float*)d_in[0];
    const float* m0  = (const float*)d_in[1];
    const float* ml  = (const float*)d_in[2];
    const float* tp  = (const float*)d_in[3];
    const float* al  = (const float*)d_in[4];
    const float* iv  = (const float*)d_in[5];
    const float* ibg = (const float*)d_in[6];
    float* out   = (float*)d_out;
    float* table = (float*)d_ws;   // B*NI*8 floats = 2 KB scratch

    constexpr size_t O1 = (size_t)B * NI * C * HW;   // g_mid_per_instance
    constexpr size_t O2 = O1 + (size_t)B * NI * HW;  // m_mid_per_instance
    constexpr size_t O3 = O2 + (size_t)B * C * HW;   // g_mid
    constexpr size_t O4 = O3 + (size_t)B * HW;       // m_mid -> then i_base

    warp_prep_kernel<<<1, 64, 0, stream>>>(ml, tp, al, iv, table);
    warp_render_kernel<<<(B * HW) / 256, 256, 0, stream>>>(
        g0, m0, ibg, table,
        out,            // g_mid_per_instance
        out + O1,       // m_mid_per_instance
        out + O2,       // g_mid
        out + O3,       // m_mid
        out + O4);      // i_base
}